// TimeSeriesStrategy_48412871361040
// MI455X (gfx1250) — compile-verified
//
#include <hip/hip_runtime.h>
#include <math.h>

// Shapes (fixed by the reference)
constexpr int Bn = 8192;   // nodes
constexpr int En = 128;    // edges per node
constexpr int An = 1024;   // actions
constexpr int Sn = 32;     // state dim
constexpr int Hn = 16;     // hidden
constexpr int Dn = 16;     // embed

typedef __attribute__((ext_vector_type(2))) float v2f;
typedef __attribute__((ext_vector_type(8))) float v8f;

__device__ __forceinline__ float gelu_exact(float x) {
    return 0.5f * x * (1.0f + erff(x * 0.70710678118654752440f));
}

// VALU cross-lane helpers (v_permlane16_b32 / v_permlanex16_b32):
// xor-butterfly within each 16-lane row, and row swap (lane ^ 16).
__device__ __forceinline__ float perml16(float x, unsigned s1, unsigned s2) {
    return __int_as_float(__builtin_amdgcn_permlane16(
        __float_as_int(x), __float_as_int(x), s1, s2, false, false));
}
__device__ __forceinline__ float permlx16_swap(float x) {
    return __int_as_float(__builtin_amdgcn_permlanex16(
        __float_as_int(x), __float_as_int(x), 0x76543210u, 0xFEDCBA98u, false, false));
}

// ---------------------------------------------------------------------------
// Kernel 1: edge encoder + mean aggregation over E, WMMA formulation.
//
//   edge_t[b] = b_e2 + W_e2 @ mean_e( relu( LN( ef[b,e] @ W_e1^T + b_e1 ) ) )
//
// (mean commutes with the 2nd linear layer -> hoisted out of the edge loop)
//
// grid = Bn blocks, block = 128 threads (4 waves); each wave processes
// 2 tiles of 16 edges via V_WMMA_F32_16X16X4_F32 (K=7 padded to 8).
// C/D layout: VGPR r, lanes 0-15 -> edge M=r, hidden N=lane;
//             lanes 16-31 -> edge M=r+8, hidden N=lane-16.
// LayerNorm over the 16 hidden units: one-pass (sum, sumsq) xor-butterfly
// on the VALU pipe (permlane16), co-executing with the WMMA/matrix pipe.
// All branches around the WMMAs are wave-uniform (EXEC all-1s).
// ---------------------------------------------------------------------------
__global__ __launch_bounds__(128) void edge_encode_wmma_kernel(
    const float* __restrict__ ef,      // [B,E,7]
    const float* __restrict__ W_e1,    // [16,7]
    const float* __restrict__ b_e1,    // [16]
    const float* __restrict__ g_e1,    // [16]
    const float* __restrict__ be_e1,   // [16]
    const float* __restrict__ W_e2,    // [16,16]
    const float* __restrict__ b_e2,    // [16]
    float* __restrict__ edge_t)        // [B,16] (workspace)
{
    __shared__ float spart[64];   // per-wave partial sums [wave][n]
    __shared__ float smh[16];     // mean of activated hidden

    const int t     = threadIdx.x;
    const int b     = blockIdx.x;
    const int lane  = t & 31;
    const int wave  = t >> 5;
    const int n     = lane & 15;            // hidden unit index
    const int khalf = (lane >> 4) << 1;     // 0 or 2 (ISA A/B layout)

    // Loop-invariant B fragments: B[k][n] = W_e1[n][k], K=7 padded with 0.
    const float* wrow = W_e1 + n * 7;
    v2f bf0, bf1;
    bf0.x = wrow[khalf];
    bf0.y = wrow[khalf + 1];
    bf1.x = (khalf == 0) ? wrow[4] : wrow[6];   // K=4 / K=6
    bf1.y = (khalf == 0) ? wrow[5] : 0.0f;      // K=5 / K=7 (pad)
    const float biasn = b_e1[n];
    const float gn    = g_e1[n];
    const float ben   = be_e1[n];

    float acc = 0.0f;   // sum over this wave's edges of relu(ln(h))[n]
#pragma unroll
    for (int tile = 0; tile < 2; ++tile) {
        // A fragments: 16 edges x 8 (K=7 padded); edge M = n for both halves.
        const int e = wave * 32 + tile * 16 + n;
        const float* p = ef + ((size_t)b * En + e) * 7;
        v2f af0, af1;
        af0.x = p[khalf];
        af0.y = p[khalf + 1];
        af1.x = (khalf == 0) ? p[4] : p[6];
        af1.y = (khalf == 0) ? p[5] : 0.0f;     // K=7 pad (also avoids OOB)

        // C init with per-column bias b_e1[n]
        v8f c;
#pragma unroll
        for (int r = 0; r < 8; ++r) c[r] = biasn;
        c = __builtin_amdgcn_wmma_f32_16x16x4_f32(false, af0, false, bf0, (short)0, c, false, false);
        c = __builtin_amdgcn_wmma_f32_16x16x4_f32(false, af1, false, bf1, (short)0, c, false, false);

        // LayerNorm over hidden dim (across each 16-lane row) + ReLU.
        // One-pass moments: butterfly (sum, sumsq) together on the VALU pipe.
#pragma unroll
        for (int r = 0; r < 8; ++r) {
            const float x = c[r];
            float s  = x;
            float ss = x * x;
            s += perml16(s, 0x67452301u, 0xEFCDAB89u);   // xor 1
            ss += perml16(ss, 0x67452301u, 0xEFCDAB89u);
            s += perml16(s, 0x45670123u, 0xCDEF89ABu);   // xor 2
            ss += perml16(ss, 0x45670123u, 0xCDEF89ABu);
            s += perml16(s, 0x32107654u, 0xBA98FEDCu);   // xor 4
            ss += perml16(ss, 0x32107654u, 0xBA98FEDCu);
            s += perml16(s, 0xFEDCBA98u, 0x76543210u);   // xor 8
            ss += perml16(ss, 0xFEDCBA98u, 0x76543210u);
            const float mean = s * (1.0f / 16.0f);
            const float var  = fmaxf(ss * (1.0f / 16.0f) - mean * mean, 0.0f);
            const float inv  = rsqrtf(var + 1e-5f);
            const float y = fmaf((x - mean) * inv, gn, ben);
            acc += fmaxf(y, 0.0f);
        }
    }
    // Fold the two 16-lane rows (edges M=0..7 vs M=8..15)
    acc += permlx16_swap(acc);
    if (lane < 16) spart[wave * 16 + lane] = acc;
    __syncthreads();
    if (t < 16)
        smh[t] = (spart[t] + spart[16 + t] + spart[32 + t] + spart[48 + t])
                 * (1.0f / (float)En);
    __syncthreads();
    // Hoisted Linear(16->16): edge_t = b_e2 + W_e2 @ mean_hidden
    if (t < 16) {
        float a = b_e2[t];
        const float* w2 = W_e2 + t * 16;
#pragma unroll
        for (int i = 0; i < 16; ++i) a = fmaf(w2[i], smh[i], a);
        edge_t[(size_t)b * 16 + t] = a;
    }
}

// ---------------------------------------------------------------------------
// Kernel 2: GRU cell + query-token MLP + wait-token head. One thread per node.
// Writes q[b] to workspace and the wait score into out[b][A].
// ---------------------------------------------------------------------------
__device__ __forceinline__ void ln16(float* x, const float* g, const float* be) {
    float m = 0.f;
#pragma unroll
    for (int i = 0; i < 16; ++i) m += x[i];
    m *= (1.0f / 16.0f);
    float v = 0.f;
#pragma unroll
    for (int i = 0; i < 16; ++i) { float d = x[i] - m; v = fmaf(d, d, v); }
    v *= (1.0f / 16.0f);
    const float inv = rsqrtf(v + 1e-5f);
#pragma unroll
    for (int i = 0; i < 16; ++i) x[i] = (x[i] - m) * inv * g[i] + be[i];
}

__global__ __launch_bounds__(256) void head_kernel(
    const float* __restrict__ edge_t,  // [B,16]
    const float* __restrict__ state,   // [B,32]
    const float* __restrict__ w_ih,    // [48,16]
    const float* __restrict__ w_hh,    // [48,16]
    const float* __restrict__ b_ih,    // [48]
    const float* __restrict__ b_hh,    // [48]
    const float* __restrict__ w_s2e,   // [16,32]
    const float* __restrict__ W_w1,    // [16,16]
    const float* __restrict__ b_w1,    // [16]
    const float* __restrict__ W_w2,    // [16,16]
    const float* __restrict__ b_w2,    // [16]
    const float* __restrict__ W_q1,    // [16,32]
    const float* __restrict__ b_q1,    // [16]
    const float* __restrict__ g_q1,    // [16]
    const float* __restrict__ be_q1,   // [16]
    const float* __restrict__ W_q2,    // [16,16]
    const float* __restrict__ b_q2,    // [16]
    const float* __restrict__ g_q2,    // [16]
    const float* __restrict__ be_q2,   // [16]
    const float* __restrict__ W_q3,    // [16,16]
    const float* __restrict__ b_q3,    // [16]
    float* __restrict__ qout,          // [B,16] (workspace)
    float* __restrict__ out)           // [B,A+1]
{
    const int b = blockIdx.x * blockDim.x + threadIdx.x;
    if (b >= Bn) return;

    float et[16], st[32];
#pragma unroll
    for (int d = 0; d < 16; ++d) et[d] = edge_t[(size_t)b * 16 + d];
#pragma unroll
    for (int s = 0; s < 32; ++s) st[s] = state[(size_t)b * 32 + s];

    // h0 = state @ w_s2e.T
    float h0[16];
#pragma unroll
    for (int d = 0; d < 16; ++d) {
        float a = 0.f;
#pragma unroll
        for (int s = 0; s < 32; ++s) a = fmaf(st[s], w_s2e[d * 32 + s], a);
        h0[d] = a;
    }
    // GRU gates
    float gi[48], gh[48];
#pragma unroll
    for (int k = 0; k < 48; ++k) {
        float a = b_ih[k], c = b_hh[k];
#pragma unroll
        for (int d = 0; d < 16; ++d) {
            a = fmaf(et[d], w_ih[k * 16 + d], a);
            c = fmaf(h0[d], w_hh[k * 16 + d], c);
        }
        gi[k] = a; gh[k] = c;
    }
    float h_t[16];
#pragma unroll
    for (int i = 0; i < 16; ++i) {
        float r = 1.0f / (1.0f + expf(-(gi[i] + gh[i])));
        float z = 1.0f / (1.0f + expf(-(gi[16 + i] + gh[16 + i])));
        float n = tanhf(gi[32 + i] + r * gh[32 + i]);
        h_t[i] = (1.0f - z) * n + z * h0[i];
    }
    // query token MLP on cat([edge_t, h_t])
    float qc[32];
#pragma unroll
    for (int i = 0; i < 16; ++i) { qc[i] = et[i]; qc[16 + i] = h_t[i]; }
    float t1[16];
#pragma unroll
    for (int i = 0; i < 16; ++i) {
        float a = b_q1[i];
#pragma unroll
        for (int j = 0; j < 32; ++j) a = fmaf(qc[j], W_q1[i * 32 + j], a);
        t1[i] = a;
    }
    ln16(t1, g_q1, be_q1);
#pragma unroll
    for (int i = 0; i < 16; ++i) t1[i] = gelu_exact(t1[i]);
    float t2[16];
#pragma unroll
    for (int i = 0; i < 16; ++i) {
        float a = b_q2[i];
#pragma unroll
        for (int j = 0; j < 16; ++j) a = fmaf(t1[j], W_q2[i * 16 + j], a);
        t2[i] = a;
    }
    ln16(t2, g_q2, be_q2);
#pragma unroll
    for (int i = 0; i < 16; ++i) t2[i] = gelu_exact(t2[i]);
    float q3[16];
#pragma unroll
    for (int d = 0; d < 16; ++d) {
        float a = b_q3[d];
#pragma unroll
        for (int j = 0; j < 16; ++j) a = fmaf(t2[j], W_q3[d * 16 + j], a);
        q3[d] = a;
        qout[(size_t)b * 16 + d] = a;
    }
    // wait token + wait score
    float u[16];
#pragma unroll
    for (int i = 0; i < 16; ++i) {
        float a = b_w1[i];
#pragma unroll
        for (int d = 0; d < 16; ++d) a = fmaf(h_t[d], W_w1[i * 16 + d], a);
        u[i] = gelu_exact(a);
    }
    float wscore = 0.f;
#pragma unroll
    for (int d = 0; d < 16; ++d) {
        float a = b_w2[d];
#pragma unroll
        for (int i = 0; i < 16; ++i) a = fmaf(u[i], W_w2[d * 16 + i], a);
        wscore = fmaf(a, q3[d], wscore);
    }
    out[(size_t)b * (An + 1) + An] = wscore;
}

// ---------------------------------------------------------------------------
// Kernel 3: scores = q @ inode_table.T with V_WMMA_F32_16X16X4_F32 + mask.
// grid = B/16 blocks of 256 threads (8 waves); wave w covers col tiles
// w, w+8, ..., 63. A fragments (q rows) loaded once per wave and reused.
// EXEC is all-1s everywhere around the WMMAs (no divergence).
// ---------------------------------------------------------------------------
__global__ __launch_bounds__(256) void scores_wmma_kernel(
    const float* __restrict__ q,     // [B,16]
    const float* __restrict__ tab,   // [A,16]
    const int*   __restrict__ avail, // [B,A]
    float* __restrict__ out)         // [B,A+1]
{
    const int lane  = threadIdx.x & 31;
    const int wave  = threadIdx.x >> 5;
    const int b0    = blockIdx.x << 4;          // 16 output rows
    const int mrow  = lane & 15;
    const int khalf = (lane >> 4) << 1;         // 0 or 2 (ISA A-layout)

    // A fragments: q rows b0..b0+15, K split into 4 steps of 4
    const float* qrow = q + (size_t)(b0 + mrow) * 16;
    v2f a0, a1, a2, a3;
    a0.x = qrow[ 0 + khalf]; a0.y = qrow[ 1 + khalf];
    a1.x = qrow[ 4 + khalf]; a1.y = qrow[ 5 + khalf];
    a2.x = qrow[ 8 + khalf]; a2.y = qrow[ 9 + khalf];
    a3.x = qrow[12 + khalf]; a3.y = qrow[13 + khalf];

    for (int ct = wave; ct < (An >> 4); ct += 8) {
        const int c0 = ct << 4;
        // B fragments: inode_table rows c0..c0+15 give columns (B = tab^T)
        const float* trow = tab + (size_t)(c0 + mrow) * 16;
        v2f bm0, bm1, bm2, bm3;
        bm0.x = trow[ 0 + khalf]; bm0.y = trow[ 1 + khalf];
        bm1.x = trow[ 4 + khalf]; bm1.y = trow[ 5 + khalf];
        bm2.x = trow[ 8 + khalf]; bm2.y = trow[ 9 + khalf];
        bm3.x = trow[12 + khalf]; bm3.y = trow[13 + khalf];

        v8f c = {0.f, 0.f, 0.f, 0.f, 0.f, 0.f, 0.f, 0.f};
        c = __builtin_amdgcn_wmma_f32_16x16x4_f32(false, a0, false, bm0, (short)0, c, false, false);
        c = __builtin_amdgcn_wmma_f32_16x16x4_f32(false, a1, false, bm1, (short)0, c, false, false);
        c = __builtin_amdgcn_wmma_f32_16x16x4_f32(false, a2, false, bm2, (short)0, c, false, false);
        c = __builtin_amdgcn_wmma_f32_16x16x4_f32(false, a3, false, bm3, (short)0, c, false, false);

        // D layout: VGPR r -> rows M=r (lanes 0-15) / M=r+8 (lanes 16-31)
        const int rbase = b0 + ((lane >> 4) << 3);
        const int col   = c0 + mrow;
        const float NEG_INF = -__builtin_inff();
#pragma unroll
        for (int r = 0; r < 8; ++r) {
            const int row = rbase + r;
            const int av  = avail[(size_t)row * An + col];
            const float val = (av > 0) ? c[r] : NEG_INF;
            out[(size_t)row * (An + 1) + col] = val;
        }
    }
}

// ---------------------------------------------------------------------------
extern "C" void kernel_launch(void* const* d_in, const int* in_sizes, int n_in,
                              void* d_out, int out_size, void* d_ws, size_t ws_size,
                              hipStream_t stream) {
    const float* edge_feats = (const float*)d_in[0];
    const float* state      = (const float*)d_in[1];
    const float* inode_tab  = (const float*)d_in[2];
    const int*   avail      = (const int*)d_in[3];
    const float* W_e1  = (const float*)d_in[4];
    const float* b_e1  = (const float*)d_in[5];
    const float* g_e1  = (const float*)d_in[6];
    const float* be_e1 = (const float*)d_in[7];
    const float* W_e2  = (const float*)d_in[8];
    const float* b_e2  = (const float*)d_in[9];
    const float* w_ih  = (const float*)d_in[10];
    const float* w_hh  = (const float*)d_in[11];
    const float* b_ih  = (const float*)d_in[12];
    const float* b_hh  = (const float*)d_in[13];
    const float* w_s2e = (const float*)d_in[14];
    // d_in[15..17]: w_e2s, W_mask, b_mask -> unused by the reference
    const float* W_w1  = (const float*)d_in[18];
    const float* b_w1  = (const float*)d_in[19];
    const float* W_w2  = (const float*)d_in[20];
    const float* b_w2  = (const float*)d_in[21];
    const float* W_q1  = (const float*)d_in[22];
    const float* b_q1  = (const float*)d_in[23];
    const float* g_q1  = (const float*)d_in[24];
    const float* be_q1 = (const float*)d_in[25];
    const float* W_q2  = (const float*)d_in[26];
    const float* b_q2  = (const float*)d_in[27];
    const float* g_q2  = (const float*)d_in[28];
    const float* be_q2 = (const float*)d_in[29];
    const float* W_q3  = (const float*)d_in[30];
    const float* b_q3  = (const float*)d_in[31];

    float* out    = (float*)d_out;
    float* edge_t = (float*)d_ws;                 // [B,16]
    float* qbuf   = edge_t + (size_t)Bn * Dn;     // [B,16]

    edge_encode_wmma_kernel<<<Bn, 128, 0, stream>>>(
        edge_feats, W_e1, b_e1, g_e1, be_e1, W_e2, b_e2, edge_t);

    head_kernel<<<Bn / 256, 256, 0, stream>>>(
        edge_t, state, w_ih, w_hh, b_ih, b_hh, w_s2e,
        W_w1, b_w1, W_w2, b_w2,
        W_q1, b_q1, g_q1, be_q1, W_q2, b_q2, g_q2, be_q2, W_q3, b_q3,
        qbuf, out);

    scores_wmma_kernel<<<Bn / 16, 256, 0, stream>>>(
        qbuf, inode_tab, avail, out);
}